// SGPMixer_38757784879896
// MI455X (gfx1250) — compile-verified
//
#include <hip/hip_runtime.h>
#include <cstdint>

#define DIMC 512
#define NB   32
#define TT   2048
#define NROW (NB * DIMC)     // 16384 rows of 2048 floats
#define EPSF 1e-5f

// ---------------------------------------------------------------------------
// Async global->LDS staging of one 2048-float row (8 KB) per block.
// Each of 256 threads issues two GLOBAL_LOAD_ASYNC_TO_LDS_B128 (16B each).
// IOFFSET is added to BOTH the LDS and global address per the ISA, so one
// base pair covers both halves. Tracked by ASYNCcnt; wave waits, then the
// block barrier makes all waves' data visible.
// NOTE: low 32 bits of a generic LDS pointer == workgroup-relative LDS byte
// offset (flat LDS addresses are {SHARED_BASE, offset[31:0]} per ISA 10.2).
// ---------------------------------------------------------------------------
__device__ __forceinline__ void stage_row_async(const float* __restrict__ grow,
                                                float* smem, int tid) {
  uint32_t l0 = (uint32_t)(uintptr_t)smem + (uint32_t)tid * 16u;
  uint64_t g0 = (uint64_t)(uintptr_t)grow + (uint64_t)tid * 16u;
  asm volatile(
      "global_load_async_to_lds_b128 %0, %1, off\n\t"
      "global_load_async_to_lds_b128 %0, %1, off offset:4096\n\t"
      "s_wait_asynccnt 0x0"
      :: "v"(l0), "v"(g0)
      : "memory");
  __syncthreads();
}

// ---------------------------------------------------------------------------
// Pass 1: per-row sum / sum-of-squares of x  (feeds BN1 stats AND phi branch)
// ---------------------------------------------------------------------------
__global__ __launch_bounds__(256) void k_rowstats1(const float* __restrict__ x,
                                                   float* __restrict__ rs,
                                                   float* __restrict__ rq) {
  const int r = blockIdx.x;            // r = b*DIMC + c (matches memory layout)
  const int b = r / DIMC, c = r % DIMC;
  const int tid = threadIdx.x;
  const float4* row = (const float4*)(x + (size_t)r * TT);

  float s = 0.f, q = 0.f;
#pragma unroll
  for (int i = 0; i < 2; ++i) {
    float4 v = row[tid + 256 * i];
    s += v.x + v.y + v.z + v.w;
    q += v.x * v.x + v.y * v.y + v.z * v.z + v.w * v.w;
  }
  __shared__ float ss[256];
  __shared__ float sq[256];
  ss[tid] = s; sq[tid] = q;
  __syncthreads();
  for (int off = 128; off > 0; off >>= 1) {   // deterministic tree
    if (tid < off) { ss[tid] += ss[tid + off]; sq[tid] += sq[tid + off]; }
    __syncthreads();
  }
  if (tid == 0) { rs[c * NB + b] = ss[0]; rq[c * NB + b] = sq[0]; }
}

// ---------------------------------------------------------------------------
// BN1 channel reduce (one wave32 per channel; lane == batch index) + phi.
// out = a_c*x + b_c ;  phi[b,c] = relu((a_c*rowmean + b_c)*gfc_w + gfc_b)
// ---------------------------------------------------------------------------
__global__ __launch_bounds__(32) void k_chan1(const float* __restrict__ rs,
                                              const float* __restrict__ rq,
                                              const float* __restrict__ norm_g,
                                              const float* __restrict__ norm_b,
                                              const float* __restrict__ gfc_w,
                                              const float* __restrict__ gfc_b,
                                              float* __restrict__ A1,
                                              float* __restrict__ B1,
                                              float* __restrict__ phi) {
  const int c = blockIdx.x;
  const int lane = threadIdx.x;        // == b
  const float s = rs[c * NB + lane];
  const float q = rq[c * NB + lane];
  float ts = s, tq = q;
#pragma unroll
  for (int off = 16; off > 0; off >>= 1) {
    ts += __shfl_xor(ts, off, 32);
    tq += __shfl_xor(tq, off, 32);
  }
  const float invN = 1.0f / ((float)NB * (float)TT);
  const float mu   = ts * invN;
  const float var  = tq * invN - mu * mu;          // biased variance
  const float rstd = rsqrtf(var + EPSF);
  const float a    = norm_g[c] * rstd;
  const float bb   = norm_b[c] - mu * a;
  if (lane == 0) { A1[c] = a; B1[c] = bb; }
  const float rowmean = s * (1.0f / (float)TT);
  const float g = a * rowmean + bb;
  const float p = g * gfc_w[c] + gfc_b[c];
  phi[lane * DIMC + c] = fmaxf(p, 0.f);
}

// ---------------------------------------------------------------------------
// Pass 2: stats of convkw pre-activation (needs BN1). d = a*(x[t]-x[t-1]),
// d[0]=d[1]; ck_pre[t] = k0*dz(t-1)+k1*dz(t)+k2*dz(t+1)+kb (zero-padded).
// ---------------------------------------------------------------------------
__global__ __launch_bounds__(256) void k_rowstats2(const float* __restrict__ x,
                                                   const float* __restrict__ A1,
                                                   const float* __restrict__ kw,
                                                   const float* __restrict__ kb,
                                                   float* __restrict__ rs2,
                                                   float* __restrict__ rq2) {
  __shared__ __align__(16) float smem[TT];
  const int r = blockIdx.x;
  const int b = r / DIMC, c = r % DIMC;
  const int tid = threadIdx.x;
  stage_row_async(x + (size_t)r * TT, smem, tid);

  const float a  = A1[c];
  const float k0 = kw[3 * c + 0], k1 = kw[3 * c + 1], k2 = kw[3 * c + 2];
  const float kbias = kb[c];

  float s = 0.f, q = 0.f;
  const int t0 = tid * 8;
#pragma unroll
  for (int j = 0; j < 8; ++j) {
    const int t = t0 + j;
    float dm, d0, dp;                                   // dz(t-1), dz(t), dz(t+1)
    { int u = t - 1; dm = (u < 0) ? 0.f : a * (smem[(u == 0) ? 1 : u] - smem[((u == 0) ? 1 : u) - 1]); }
    { int u = t;     d0 = a * (smem[(u == 0) ? 1 : u] - smem[((u == 0) ? 1 : u) - 1]); }
    { int u = t + 1; dp = (u >= TT) ? 0.f : a * (smem[u] - smem[u - 1]); }
    const float v = k0 * dm + k1 * d0 + k2 * dp + kbias;
    s += v; q += v * v;
  }
  __syncthreads();                       // done reading smem; reduce uses separate ss/sq
  __shared__ float ss[256];
  __shared__ float sq[256];
  ss[tid] = s; sq[tid] = q;
  __syncthreads();
  for (int off = 128; off > 0; off >>= 1) {
    if (tid < off) { ss[tid] += ss[tid + off]; sq[tid] += sq[tid + off]; }
    __syncthreads();
  }
  if (tid == 0) { rs2[c * NB + b] = ss[0]; rq2[c * NB + b] = sq[0]; }
}

// ---------------------------------------------------------------------------
// BN2 channel reduce: convkw = A2*ck_pre + B2
// ---------------------------------------------------------------------------
__global__ __launch_bounds__(32) void k_chan2(const float* __restrict__ rs,
                                              const float* __restrict__ rq,
                                              const float* __restrict__ dn_g,
                                              const float* __restrict__ dn_b,
                                              float* __restrict__ A2,
                                              float* __restrict__ B2) {
  const int c = blockIdx.x;
  const int lane = threadIdx.x;
  float ts = rs[c * NB + lane], tq = rq[c * NB + lane];
#pragma unroll
  for (int off = 16; off > 0; off >>= 1) {
    ts += __shfl_xor(ts, off, 32);
    tq += __shfl_xor(tq, off, 32);
  }
  if (lane == 0) {
    const float invN = 1.0f / ((float)NB * (float)TT);
    const float mu   = ts * invN;
    const float var  = tq * invN - mu * mu;
    const float rstd = rsqrtf(var + EPSF);
    const float a2   = dn_g[c] * rstd;
    A2[c] = a2;
    B2[c] = dn_b[c] - mu * a2;
  }
}

// ---------------------------------------------------------------------------
// Pass 3: fused final. One block per row; row staged to LDS via async DMA.
// ---------------------------------------------------------------------------
__global__ __launch_bounds__(256) void k_final(
    const float* __restrict__ x,
    const float* __restrict__ A1c, const float* __restrict__ B1c,
    const float* __restrict__ phi,
    const float* __restrict__ psi_w, const float* __restrict__ psi_b,
    const float* __restrict__ fc_w,  const float* __restrict__ fc_b,
    const float* __restrict__ cw_w,  const float* __restrict__ cw_b,
    const float* __restrict__ kw_w,  const float* __restrict__ kw_b,
    const float* __restrict__ A2c,   const float* __restrict__ B2c,
    float* __restrict__ out) {
  __shared__ __align__(16) float smem[TT];
  const int r = blockIdx.x;
  const int c = r % DIMC;
  const int tid = threadIdx.x;
  stage_row_async(x + (size_t)r * TT, smem, tid);

  const float a  = A1c[c], bb = B1c[c];
  const float pw0 = psi_w[3 * c], pw1 = psi_w[3 * c + 1], pw2 = psi_w[3 * c + 2];
  const float pbias = psi_b[c];
  const float cw0 = cw_w[3 * c], cw1 = cw_w[3 * c + 1], cw2 = cw_w[3 * c + 2];
  const float cbias = cw_b[c];
  const float k0 = kw_w[3 * c], k1 = kw_w[3 * c + 1], k2 = kw_w[3 * c + 2];
  const float kbias = kw_b[c];
  const float fw = fc_w[c], fb = fc_b[c];
  const float a2 = A2c[c], b2 = B2c[c];
  const float ph = phi[r];                    // r == b*DIMC + c

  const int t0 = tid * 8;
  float res[8];
#pragma unroll
  for (int j = 0; j < 8; ++j) {
    const int t = t0 + j;
    const float xt = smem[t];
    const float o  = a * xt + bb;
    const float om = (t > 0)      ? a * smem[t - 1] + bb : 0.f;   // zero-pad conv input
    const float op = (t < TT - 1) ? a * smem[t + 1] + bb : 0.f;
    const float psi = pw0 * om + pw1 * o + pw2 * op + pbias;
    const float cvw = cw0 * om + cw1 * o + cw2 * op + cbias;
    float dm, d0, dp;
    { int u = t - 1; dm = (u < 0) ? 0.f : a * (smem[(u == 0) ? 1 : u] - smem[((u == 0) ? 1 : u) - 1]); }
    { int u = t;     d0 = a * (smem[(u == 0) ? 1 : u] - smem[((u == 0) ? 1 : u) - 1]); }
    { int u = t + 1; dp = (u >= TT) ? 0.f : a * (smem[u] - smem[u - 1]); }
    const float ck  = k0 * dm + k1 * d0 + k2 * dp + kbias;
    const float ckw = a2 * ck + b2;
    const float fc  = o * fw + fb;
    res[j] = xt + fc * ph + (cvw + ckw) * psi + o;
  }
  float4* orow = (float4*)(out + (size_t)r * TT + t0);
  orow[0] = make_float4(res[0], res[1], res[2], res[3]);
  orow[1] = make_float4(res[4], res[5], res[6], res[7]);
}

// ---------------------------------------------------------------------------
extern "C" void kernel_launch(void* const* d_in, const int* in_sizes, int n_in,
                              void* d_out, int out_size, void* d_ws, size_t ws_size,
                              hipStream_t stream) {
  (void)in_sizes; (void)n_in; (void)out_size; (void)ws_size;
  const float* x       = (const float*)d_in[0];
  const float* norm_g  = (const float*)d_in[1];
  const float* norm_b  = (const float*)d_in[2];
  const float* psi_w   = (const float*)d_in[3];
  const float* psi_b   = (const float*)d_in[4];
  const float* fc_w    = (const float*)d_in[5];
  const float* fc_b    = (const float*)d_in[6];
  const float* convw_w = (const float*)d_in[7];
  const float* convw_b = (const float*)d_in[8];
  const float* convkw_w= (const float*)d_in[9];
  const float* convkw_b= (const float*)d_in[10];
  const float* dn_g    = (const float*)d_in[11];
  const float* dn_b    = (const float*)d_in[12];
  const float* gfc_w   = (const float*)d_in[13];
  const float* gfc_b   = (const float*)d_in[14];
  float* out = (float*)d_out;
  float* ws  = (float*)d_ws;

  float* rs1 = ws;                               // [NROW] idx c*NB+b
  float* rq1 = ws + NROW;                        // [NROW]
  float* A1  = ws + 2 * NROW;                    // [DIMC]
  float* B1  = ws + 2 * NROW + DIMC;             // [DIMC]
  float* phi = ws + 2 * NROW + 2 * DIMC;         // [NROW] idx b*DIMC+c
  float* rs2 = ws + 3 * NROW + 2 * DIMC;         // [NROW]
  float* rq2 = ws + 4 * NROW + 2 * DIMC;         // [NROW]
  float* A2  = ws + 5 * NROW + 2 * DIMC;         // [DIMC]
  float* B2  = ws + 5 * NROW + 3 * DIMC;         // [DIMC]

  k_rowstats1<<<NROW, 256, 0, stream>>>(x, rs1, rq1);
  k_chan1<<<DIMC, 32, 0, stream>>>(rs1, rq1, norm_g, norm_b, gfc_w, gfc_b, A1, B1, phi);
  k_rowstats2<<<NROW, 256, 0, stream>>>(x, A1, convkw_w, convkw_b, rs2, rq2);
  k_chan2<<<DIMC, 32, 0, stream>>>(rs2, rq2, dn_g, dn_b, A2, B2);
  k_final<<<NROW, 256, 0, stream>>>(x, A1, B1, phi, psi_w, psi_b, fc_w, fc_b,
                                    convw_w, convw_b, convkw_w, convkw_b, A2, B2, out);
}